// KNNAttention_3539053052746
// MI455X (gfx1250) — compile-verified
//
#include <hip/hip_runtime.h>
#include <hip/hip_bf16.h>
#include <math.h>

// Shapes fixed by the reference: b=4, l=2048, d=1024, h=16, dh=64
#define BATCH 4
#define LSEQ  2048
#define DMODEL 1024
#define NHEAD 16
#define DHEAD 64

typedef __attribute__((ext_vector_type(2))) float v2f;
typedef __attribute__((ext_vector_type(8))) float v8f;

__device__ __forceinline__ v8f wmma_f32(v2f a, v2f b, v8f c) {
  // D(16x16,f32) = A(16x4,f32) x B(4x16,f32) + C
  return __builtin_amdgcn_wmma_f32_16x16x4_f32(
      /*neg_a=*/false, a, /*neg_b=*/false, b,
      /*c_mod=*/(short)0, c, /*reuse_a=*/false, /*reuse_b=*/false);
}

// ---------------------------------------------------------------------------
// C[8192,1024] = A[8192,1024] * W[1024,1024]^T  (torch Linear layout).
// One 16x128 strip per wave; all fragment loads use immediate offsets so the
// compiler can clause them (compile-time strides, j*64KB < 8MB IOFFSET range).
// ---------------------------------------------------------------------------
__global__ void gemm_nt_kernel(const float* __restrict__ A,
                               const float* __restrict__ W,
                               float* __restrict__ C) {
  const int lane = threadIdx.x & 31;
  const int wave = threadIdx.x >> 5;
  const int Ntb = DMODEL >> 7;                  // 8 column blocks of 128
  const int t = blockIdx.x * (blockDim.x >> 5) + wave;
  if (t >= (BATCH * LSEQ / 16) * Ntb) return;
  const int m0 = (t / Ntb) << 4;
  const int n0 = (t % Ntb) << 7;
  const int lo = lane & 15, hi = lane >> 4;

  const float* arow = A + (size_t)(m0 + lo) * DMODEL;
  const float* wrow = W + (size_t)(n0 + lo) * DMODEL;

  v8f acc[8] = {};
  for (int k = 0; k < DMODEL; k += 4) {
    const int kk = k + 2 * hi;
    if ((k & 63) == 0) __builtin_prefetch(arow + k + 256, 0, 3);
    v2f a = { arow[kk], arow[kk + 1] };
#pragma unroll
    for (int j = 0; j < 8; ++j) {
      v2f b = { wrow[kk + j * 16 * DMODEL], wrow[kk + j * 16 * DMODEL + 1] };
      acc[j] = wmma_f32(a, b, acc[j]);
    }
  }
  float* crow = C + (size_t)(m0 + 8 * hi) * DMODEL + n0 + lo;
#pragma unroll
  for (int j = 0; j < 8; ++j)
#pragma unroll
    for (int r = 0; r < 8; ++r)
      crow[r * DMODEL + j * 16] = acc[j][r];
}

// ---------------------------------------------------------------------------
// Top-1 kNN: scores[q,m] = qp[b,q,:] . kv[b,m,:]; idx[b,q] = argmax_m (first max)
// One wave per (b, 16-query tile); processes m-tiles in groups of 4 so each
// A fragment feeds 4 WMMAs.
// ---------------------------------------------------------------------------
__global__ void knn_argmax_kernel(const float* __restrict__ qp,
                                  const float* __restrict__ kv,
                                  int* __restrict__ idx) {
  const int lane = threadIdx.x & 31;
  const int wave = threadIdx.x >> 5;
  const int t = blockIdx.x * (blockDim.x >> 5) + wave;   // BATCH*128 tasks
  if (t >= BATCH * (LSEQ / 16)) return;
  const int b  = t >> 7;
  const int q0 = (t & 127) << 4;
  const int lo = lane & 15, hi = lane >> 4;

  const float* qpb = qp + (size_t)b * LSEQ * DMODEL;
  const float* kvb = kv + (size_t)b * LSEQ * DMODEL;
  const float* arow = qpb + (size_t)(q0 + lo) * DMODEL;

  float bestv[8];
  int   bestm[8];
#pragma unroll
  for (int r = 0; r < 8; ++r) { bestv[r] = -INFINITY; bestm[r] = 0; }

  for (int g = 0; g < LSEQ / 64; ++g) {        // 32 groups of 4 m-tiles
    const int m0 = g << 6;
    const float* brow0 = kvb + (size_t)(m0 + lo) * DMODEL;
    v8f c[4] = {};
    for (int k = 0; k < DMODEL; k += 4) {
      const int kk = k + 2 * hi;
      if ((k & 63) == 0) __builtin_prefetch(arow + k + 256, 0, 3);
      v2f a = { arow[kk], arow[kk + 1] };
#pragma unroll
      for (int j = 0; j < 4; ++j) {
        v2f bb = { brow0[kk + j * 16 * DMODEL], brow0[kk + j * 16 * DMODEL + 1] };
        c[j] = wmma_f32(a, bb, c[j]);
      }
    }
#pragma unroll
    for (int j = 0; j < 4; ++j) {
      const int mcol = m0 + j * 16 + lo;       // ascending m => strict > keeps first max
#pragma unroll
      for (int r = 0; r < 8; ++r) {
        const float v = c[j][r];
        if (v > bestv[r]) { bestv[r] = v; bestm[r] = mcol; }
      }
    }
  }
  // butterfly max across the 16 lanes of each column group (same hi)
#pragma unroll
  for (int off = 1; off < 16; off <<= 1) {
#pragma unroll
    for (int r = 0; r < 8; ++r) {
      const float ov = __shfl_xor(bestv[r], off, 32);
      const int   om = __shfl_xor(bestm[r], off, 32);
      if (ov > bestv[r] || (ov == bestv[r] && om < bestm[r])) {
        bestv[r] = ov; bestm[r] = om;
      }
    }
  }
  if (lo == 0) {
#pragma unroll
    for (int r = 0; r < 8; ++r)
      idx[(size_t)b * LSEQ + q0 + 8 * hi + r] = bestm[r];
  }
}

// ---------------------------------------------------------------------------
// mkv[b,q,0:128] = kv[b, idx[b,q], :] @ Wkv[128,1024]^T   (gathered-row GEMM)
// One wave per (b, qtile): full 16x128 strip, A fragment shared by 8 WMMAs.
// ---------------------------------------------------------------------------
__global__ void gather_proj_kernel(const float* __restrict__ kv,
                                   const int* __restrict__ idx,
                                   const float* __restrict__ Wkv,
                                   float* __restrict__ mkv) {
  const int lane = threadIdx.x & 31;
  const int wave = threadIdx.x >> 5;
  const int t = blockIdx.x * (blockDim.x >> 5) + wave;   // BATCH*128 tasks
  if (t >= BATCH * (LSEQ / 16)) return;
  const int b  = t >> 7;
  const int q0 = (t & 127) << 4;
  const int lo = lane & 15, hi = lane >> 4;

  const int gq = idx[(size_t)b * LSEQ + q0 + lo];        // gathered kv row for this A row
  const float* arow  = kv + ((size_t)b * LSEQ + gq) * DMODEL;
  const float* wrow0 = Wkv + (size_t)lo * DMODEL;

  v8f acc[8] = {};
  for (int k = 0; k < DMODEL; k += 4) {
    const int kk = k + 2 * hi;
    v2f a = { arow[kk], arow[kk + 1] };
#pragma unroll
    for (int j = 0; j < 8; ++j) {
      v2f b = { wrow0[kk + j * 16 * DMODEL], wrow0[kk + j * 16 * DMODEL + 1] };
      acc[j] = wmma_f32(a, b, acc[j]);
    }
  }
  float* crow = mkv + ((size_t)b * LSEQ + q0 + 8 * hi) * 128 + lo;
#pragma unroll
  for (int j = 0; j < 8; ++j)
#pragma unroll
    for (int r = 0; r < 8; ++r)
      crow[r * 128 + j * 16] = acc[j][r];
}

// ---------------------------------------------------------------------------
// Flash attention: per (b,h,qtile) wave, online softmax over 128 m-tiles.
// mk = mkv[...,0:64], mv = mkv[...,64:128]. aout[b,q, h*64 + n].
// ---------------------------------------------------------------------------
__global__ void attention_kernel(const float* __restrict__ qp,
                                 const float* __restrict__ mkv,
                                 float* __restrict__ aout) {
  __shared__ float Pt[8][16][17];                        // per-wave P tile (padded)
  const int lane = threadIdx.x & 31;
  const int wave = threadIdx.x >> 5;
  int t = blockIdx.x * (blockDim.x >> 5) + wave;         // BATCH*NHEAD*128 tasks (exact grid)
  const int q0 = (t & 127) << 4;  t >>= 7;
  const int h  = t & 15;
  const int b  = t >> 4;
  const int lo = lane & 15, hi = lane >> 4;

  const float* qpb  = qp  + (size_t)b * LSEQ * DMODEL;
  const float* mkvb = mkv + (size_t)b * LSEQ * 128;

  // Preload q A-fragments for K = 64 (16 chunks of 4)
  v2f aq[16];
  const float* arow = qpb + (size_t)(q0 + lo) * DMODEL + h * DHEAD;
#pragma unroll
  for (int kc = 0; kc < 16; ++kc) {
    const int kk = kc * 4 + 2 * hi;
    aq[kc].x = arow[kk];
    aq[kc].y = arow[kk + 1];
  }

  float mi[8], li[8];
  v8f acc[4] = {};
#pragma unroll
  for (int r = 0; r < 8; ++r) { mi[r] = -INFINITY; li[r] = 0.f; }
  const float scale = 0.125f;   // 1/sqrt(64)

  for (int mt = 0; mt < LSEQ / 16; ++mt) {
    const int m0 = mt << 4;
    // S tile = q . mk^T
    v8f s = {};
    const float* brow = mkvb + (size_t)(m0 + lo) * 128;  // mk columns 0..63
#pragma unroll
    for (int kc = 0; kc < 16; ++kc) {
      const int kk = kc * 4 + 2 * hi;
      v2f bb = { brow[kk], brow[kk + 1] };
      s = wmma_f32(aq[kc], bb, s);
    }
    float sv[8], rm[8];
#pragma unroll
    for (int r = 0; r < 8; ++r) { sv[r] = s[r] * scale; rm[r] = sv[r]; }
    // row max across 16-lane column group
#pragma unroll
    for (int off = 1; off < 16; off <<= 1)
#pragma unroll
      for (int r = 0; r < 8; ++r) rm[r] = fmaxf(rm[r], __shfl_xor(rm[r], off, 32));

    float p[8], alpha[8], rs[8];
#pragma unroll
    for (int r = 0; r < 8; ++r) {
      const float mn = fmaxf(mi[r], rm[r]);
      alpha[r] = __expf(mi[r] - mn);
      mi[r] = mn;
      p[r]  = __expf(sv[r] - mn);
      rs[r] = p[r];
    }
#pragma unroll
    for (int off = 1; off < 16; off <<= 1)
#pragma unroll
      for (int r = 0; r < 8; ++r) rs[r] += __shfl_xor(rs[r], off, 32);
#pragma unroll
    for (int r = 0; r < 8; ++r) li[r] = li[r] * alpha[r] + rs[r];
#pragma unroll
    for (int nt = 0; nt < 4; ++nt)
#pragma unroll
      for (int r = 0; r < 8; ++r) acc[nt][r] *= alpha[r];

    // transpose P (C layout -> A layout) through LDS
    __syncthreads();
#pragma unroll
    for (int r = 0; r < 8; ++r) Pt[wave][r + 8 * hi][lo] = p[r];
    __syncthreads();
    v2f ap[4];
#pragma unroll
    for (int kc = 0; kc < 4; ++kc) {
      const int kk = kc * 4 + 2 * hi;
      ap[kc].x = Pt[wave][lo][kk];
      ap[kc].y = Pt[wave][lo][kk + 1];
    }
    // O += P . mv_tile   (mv rows m0+kk, columns nt*16+lo of dh)
#pragma unroll
    for (int nt = 0; nt < 4; ++nt) {
#pragma unroll
      for (int kc = 0; kc < 4; ++kc) {
        const int kk = kc * 4 + 2 * hi;
        const float* mvp = mkvb + (size_t)(m0 + kk) * 128 + 64 + nt * 16 + lo;
        v2f bb = { mvp[0], mvp[128] };
        acc[nt] = wmma_f32(ap[kc], bb, acc[nt]);
      }
    }
  }

  // normalize and store: aout[b, q, h*64 + n]
  float* orow = aout + ((size_t)b * LSEQ + q0) * DMODEL + h * DHEAD;
#pragma unroll
  for (int r = 0; r < 8; ++r) {
    const float inv = 1.0f / li[r];
    const int row = r + 8 * hi;
#pragma unroll
    for (int nt = 0; nt < 4; ++nt)
      orow[(size_t)row * DMODEL + nt * 16 + lo] = acc[nt][r] * inv;
  }
}

// ---------------------------------------------------------------------------
extern "C" void kernel_launch(void* const* d_in, const int* in_sizes, int n_in,
                              void* d_out, int out_size, void* d_ws, size_t ws_size,
                              hipStream_t stream) {
  (void)in_sizes; (void)n_in; (void)out_size; (void)ws_size;
  const float* q   = (const float*)d_in[0];
  const float* kv  = (const float*)d_in[1];
  const float* Wq  = (const float*)d_in[2];
  const float* Wkv = (const float*)d_in[3];
  const float* Wc  = (const float*)d_in[4];
  float* out = (float*)d_out;

  const int M = BATCH * LSEQ;                 // 8192
  float* qp   = (float*)d_ws;                 // [8192,1024]
  float* aout = qp + (size_t)M * DMODEL;      // [8192,1024]
  float* mkv  = aout + (size_t)M * DMODEL;    // [8192,128]
  int*   idx  = (int*)(mkv + (size_t)M * 128);// [8192]

  const int threads = 256;                    // 8 waves (wave32)
  const int wavesPerBlock = threads / 32;

  // 1) qp = q @ Wq^T  (16x128 strip per wave)
  {
    const int tasks = (M / 16) * (DMODEL / 128);         // 4096
    gemm_nt_kernel<<<tasks / wavesPerBlock, threads, 0, stream>>>(q, Wq, qp);
  }
  // 2) top-1 kNN argmax
  {
    const int tasks = BATCH * (LSEQ / 16);               // 512
    knn_argmax_kernel<<<tasks / wavesPerBlock, threads, 0, stream>>>(qp, kv, idx);
  }
  // 3) gather + project to mk/mv
  {
    const int tasks = BATCH * (LSEQ / 16);               // 512
    gather_proj_kernel<<<tasks / wavesPerBlock, threads, 0, stream>>>(
        kv, idx, Wkv, mkv);
  }
  // 4) flash attention
  {
    const int tasks = BATCH * NHEAD * (LSEQ / 16);       // 8192
    attention_kernel<<<tasks / wavesPerBlock, threads, 0, stream>>>(qp, mkv, aout);
  }
  // 5) out = aout @ Wc^T
  {
    const int tasks = (M / 16) * (DMODEL / 128);         // 4096
    gemm_nt_kernel<<<tasks / wavesPerBlock, threads, 0, stream>>>(aout, Wc, out);
  }
}